// FastAttentionviaLowRankDecomposition_77068893159621
// MI455X (gfx1250) — compile-verified
//
#include <hip/hip_runtime.h>

// Performer (FAVOR+) causal linear attention for B=1,H=8,N=1024,D=64,M=128.
// Full-fp32 path using V_WMMA_F32_16X16X4_F32 so results track the fp32
// reference closely. One workgroup per head; causal scan chunked by 16 rows,
// all per-chunk work expressed as 16x16 WMMA tiles with S-state in LDS.

#define KEPS 1e-3f
#define NSEQ 1024
#define DH   64
#define MF   128
#define CHK  16

typedef float v2f __attribute__((ext_vector_type(2)));
typedef float v8f __attribute__((ext_vector_type(8)));

__device__ __forceinline__ v8f wmma4(v2f a, v2f b, v8f c) {
  // D = A(16x4,f32) * B(4x16,f32) + C(16x16,f32)
  return __builtin_amdgcn_wmma_f32_16x16x4_f32(false, a, false, b,
                                               (short)0, c, false, false);
}

__global__ __launch_bounds__(256)
void performer_causal_f32_wmma(const float* __restrict__ q,
                               const float* __restrict__ k,
                               const float* __restrict__ v,
                               const float* __restrict__ proj,
                               float* __restrict__ out) {
  __shared__ float proj_s[MF * DH];   // proj[m][d], row-major        32 KB
  __shared__ float S_s[MF * DH];      // running state S[m][e]        32 KB
  __shared__ float qp_s[CHK * MF];    // qp chunk, later qn in place   8 KB
  __shared__ float kp_s[CHK * MF];    // kp chunk                      8 KB
  __shared__ float xq_s[CHK * DH];    // q rows of chunk               4 KB
  __shared__ float xk_s[CHK * DH];    // k rows of chunk               4 KB
  __shared__ float v_s[CHK * DH];     // v rows of chunk               4 KB
  __shared__ float a_s[CHK * CHK];    // tril(qn @ kp^T)               1 KB
  __shared__ float kc_s[MF];          // feature cumsum carry

  const int h    = blockIdx.x;        // 8 heads -> 8 workgroups
  const int tid  = threadIdx.x;       // 256 threads = 8 wave32
  const int lane = tid & 31;
  const int wave = tid >> 5;
  const int lr   = lane & 15;         // row/col within 16
  const int lh   = lane >> 4;         // lane-half selects k / row group

  const float norm = 0.35355339059327378f;  // 64^-0.25

  const float* qh = q + (size_t)h * NSEQ * DH;
  const float* kh = k + (size_t)h * NSEQ * DH;
  const float* vh = v + (size_t)h * NSEQ * DH;
  float*       oh = out + (size_t)h * NSEQ * DH;

  // ---- init: stage proj, zero state ----
  for (int i = tid; i < MF * DH; i += 256) { proj_s[i] = proj[i]; S_s[i] = 0.f; }
  for (int i = tid; i < MF; i += 256) kc_s[i] = 0.f;
  __syncthreads();

  for (int n0 = 0; n0 < NSEQ; n0 += CHK) {
    // ---- phase 0: stage chunk inputs ----
    for (int i = tid; i < CHK * DH; i += 256) {
      xq_s[i] = qh[n0 * DH + i];
      xk_s[i] = kh[n0 * DH + i];
      v_s[i]  = vh[n0 * DH + i];
    }
    __syncthreads();

    // ---- phase 1: qp/kp feature tiles; wave w owns M cols [16w,16w+16) ----
    {
      const int m0 = wave * 16;
      v8f accq = {};
      v8f acck = {};
#pragma unroll
      for (int kb = 0; kb < DH; kb += 4) {
        v2f a, b;
        // B[k][n] = proj[m0+n][k]
        b.x = proj_s[(m0 + lr) * DH + kb + 2 * lh];
        b.y = proj_s[(m0 + lr) * DH + kb + 2 * lh + 1];
        a.x = xq_s[lr * DH + kb + 2 * lh];
        a.y = xq_s[lr * DH + kb + 2 * lh + 1];
        accq = wmma4(a, b, accq);
        a.x = xk_s[lr * DH + kb + 2 * lh];
        a.y = xk_s[lr * DH + kb + 2 * lh + 1];
        acck = wmma4(a, b, acck);
      }
#pragma unroll
      for (int j = 0; j < 8; ++j) {
        const int row = j + 8 * lh;
        qp_s[row * MF + m0 + lr] = fmaxf(accq[j] * norm, 0.f) + KEPS;
        kp_s[row * MF + m0 + lr] = fmaxf(acck[j] * norm, 0.f) + KEPS;
      }
    }
    __syncthreads();

    // ---- phase 2: per-feature causal cumsum, qn = qp / Kc (in place) ----
    if (tid < MF) {
      float run = kc_s[tid];
#pragma unroll
      for (int r = 0; r < CHK; ++r) {
        run += kp_s[r * MF + tid];
        qp_s[r * MF + tid] = qp_s[r * MF + tid] / run;
      }
      kc_s[tid] = run;
    }
    __syncthreads();

    // ---- phase 3a: out_inter = qn @ S_prev (waves 0-3), A = tril(qn@kp^T) (wave 4)
    v8f o = {};
    if (wave < 4) {
      const int e0 = wave * 16;
#pragma unroll
      for (int kb = 0; kb < MF; kb += 4) {
        v2f a, b;
        a.x = qp_s[lr * MF + kb + 2 * lh];          // qn row lr
        a.y = qp_s[lr * MF + kb + 2 * lh + 1];
        b.x = S_s[(kb + 2 * lh) * DH + e0 + lr];    // B[m][e] = S[m][e]
        b.y = S_s[(kb + 2 * lh + 1) * DH + e0 + lr];
        o = wmma4(a, b, o);
      }
    } else if (wave == 4) {
      v8f am = {};
#pragma unroll
      for (int kb = 0; kb < MF; kb += 4) {
        v2f a, b;
        a.x = qp_s[lr * MF + kb + 2 * lh];          // qn row lr
        a.y = qp_s[lr * MF + kb + 2 * lh + 1];
        b.x = kp_s[lr * MF + kb + 2 * lh];          // B[m][i] = kp[i][m], col i=lr
        b.y = kp_s[lr * MF + kb + 2 * lh + 1];
        am = wmma4(a, b, am);
      }
#pragma unroll
      for (int j = 0; j < 8; ++j) {
        const int row = j + 8 * lh;                 // output row n
        a_s[row * CHK + lr] = (lr <= row) ? am[j] : 0.f;  // causal mask i<=n
      }
    }
    __syncthreads();

    // ---- phase 3b: out = out_inter + tril(A) @ V  (waves 0-3, C kept in regs)
    if (wave < 4) {
      const int e0 = wave * 16;
#pragma unroll
      for (int kb = 0; kb < CHK; kb += 4) {
        v2f a, b;
        a.x = a_s[lr * CHK + kb + 2 * lh];
        a.y = a_s[lr * CHK + kb + 2 * lh + 1];
        b.x = v_s[(kb + 2 * lh) * DH + e0 + lr];
        b.y = v_s[(kb + 2 * lh + 1) * DH + e0 + lr];
        o = wmma4(a, b, o);
      }
#pragma unroll
      for (int j = 0; j < 8; ++j) {
        const int row = j + 8 * lh;
        oh[(n0 + row) * DH + e0 + lr] = o[j];
      }
    }
    // ---- S += kp^T @ V : wave w owns S feature rows [16w,16w+16), 4 e-tiles
    {
      const int m0 = wave * 16;
#pragma unroll
      for (int et = 0; et < 4; ++et) {
        const int e0 = et * 16;
        v8f sacc;
#pragma unroll
        for (int j = 0; j < 8; ++j)
          sacc[j] = S_s[(m0 + j + 8 * lh) * DH + e0 + lr];
#pragma unroll
        for (int kb = 0; kb < CHK; kb += 4) {
          v2f a, b;
          a.x = kp_s[(kb + 2 * lh) * MF + m0 + lr];      // A[m][i] = kp[i][m]
          a.y = kp_s[(kb + 2 * lh + 1) * MF + m0 + lr];
          b.x = v_s[(kb + 2 * lh) * DH + e0 + lr];       // B[i][e] = v[i][e]
          b.y = v_s[(kb + 2 * lh + 1) * DH + e0 + lr];
          sacc = wmma4(a, b, sacc);
        }
#pragma unroll
        for (int j = 0; j < 8; ++j)
          S_s[(m0 + j + 8 * lh) * DH + e0 + lr] = sacc[j];
      }
    }
    __syncthreads();
  }
}

extern "C" void kernel_launch(void* const* d_in, const int* in_sizes, int n_in,
                              void* d_out, int out_size, void* d_ws, size_t ws_size,
                              hipStream_t stream) {
  (void)in_sizes; (void)n_in; (void)out_size; (void)d_ws; (void)ws_size;
  const float* q    = (const float*)d_in[0];
  const float* k    = (const float*)d_in[1];
  const float* v    = (const float*)d_in[2];
  const float* proj = (const float*)d_in[3];
  float* out        = (float*)d_out;
  performer_causal_f32_wmma<<<dim3(8), dim3(256), 0, stream>>>(q, k, v, proj, out);
}